// HashEmbedder3D_88261577933508
// MI455X (gfx1250) — compile-verified
//
#include <hip/hip_runtime.h>
#include <stdint.h>

// ---------------------------------------------------------------------------
// HashEmbedder3D for MI455X (gfx1250).
// 16 levels, F=2, T=2^19. Levels 0..7 dense ((res+1)^3 rows, index stride=res
// exactly as the reference), levels 8..15 hashed (T rows).
// Memory-bound gather kernel: table (~42MB) is L2-resident (192MB L2); output
// (128MB) is streamed with non-temporal stores; level-0 table (39KB) is staged
// into LDS via the CDNA5 async global->LDS path when available.
// ---------------------------------------------------------------------------

namespace {
constexpr int kLevels = 16;
constexpr int kDense  = 8;                  // levels with res^3 < T
constexpr int kT      = 1 << 19;
constexpr int kRes[kLevels] = {16, 20, 25, 32, 40, 50, 64, 80,
                               101, 128, 161, 203, 256, 322, 406, 512};
// OFFS = cumsum of sizes: 17^3,21^3,26^3,33^3,41^3,51^3,65^3,81^3, then 8x T
constexpr int kOff[kLevels] = {0, 4913, 14174, 31750, 67687, 136608, 269259, 543884,
                               1075325, 1599613, 2123901, 2648189,
                               3172477, 3696765, 4221053, 4745341};
constexpr int kL0Rows = 4913;               // (16+1)^3 rows, 39304 B in LDS
constexpr uint32_t P1 = 2654435761u, P2 = 805459861u;
}

typedef int   v2i __attribute__((ext_vector_type(2)));
typedef float v4f __attribute__((ext_vector_type(4)));

#if __has_builtin(__builtin_amdgcn_global_load_async_to_lds_b64) && \
    __has_builtin(__builtin_amdgcn_s_wait_asynccnt)
#define HG_ASYNC_LDS 1
typedef __attribute__((address_space(1))) v2i GV2i;   // global b64 payload
typedef __attribute__((address_space(3))) v2i LV2i;   // LDS b64 payload
#else
#define HG_ASYNC_LDS 0
#endif

__device__ __forceinline__ float2 lrp(float2 a, float2 b, float w) {
  const float ow = 1.0f - w;                // matches reference a*(1-w)+b*w
  return make_float2(a.x * ow + b.x * w, a.y * ow + b.y * w);
}

__launch_bounds__(256)
__global__ void hashgrid3d_kernel(const float* __restrict__ x,
                                  const float2* __restrict__ tables,
                                  float* __restrict__ out,
                                  int npts)
{
  __shared__ float2 lut0[kL0Rows];
  const int tid = threadIdx.x;

  // ---- Stage level-0 dense table into LDS (async DMA path on CDNA5) ----
#if HG_ASYNC_LDS
  for (int i = tid; i < kL0Rows; i += 256) {
    __builtin_amdgcn_global_load_async_to_lds_b64(
        (GV2i*)(tables + i), (LV2i*)(&lut0[i]), /*offset=*/0, /*cpol=*/0);
  }
  __builtin_amdgcn_s_wait_asynccnt(0);
#else
  for (int i = tid; i < kL0Rows; i += 256) lut0[i] = tables[i];
#endif
  __syncthreads();

  const int stride = gridDim.x * blockDim.x;
  for (int p = blockIdx.x * blockDim.x + tid; p < npts; p += stride) {
    if (p + stride < npts)
      __builtin_prefetch(&x[3 * (p + stride)], 0, 0);   // global_prefetch_b8

    const float px = x[3 * p + 0];
    const float py = x[3 * p + 1];
    const float pz = x[3 * p + 2];

    float o[2 * kLevels];

    #pragma unroll
    for (int l = 0; l < kLevels; ++l) {
      const int   res  = kRes[l];
      const float grid = 2.0f / (float)res;     // folds to the f32 constant
      const float invg = (float)res * 0.5f;     // 1/grid (exact for pow2 res)

      float xc, t;
      int b0, b1, b2;
      float w0, w1, w2;
      xc = fminf(fmaxf(px, -1.0f), 1.0f); t = (xc + 1.0f) * invg; b0 = (int)t;
      w0 = (px - ((float)b0 * grid - 1.0f)) * invg;
      xc = fminf(fmaxf(py, -1.0f), 1.0f); t = (xc + 1.0f) * invg; b1 = (int)t;
      w1 = (py - ((float)b1 * grid - 1.0f)) * invg;
      xc = fminf(fmaxf(pz, -1.0f), 1.0f); t = (xc + 1.0f) * invg; b2 = (int)t;
      w2 = (pz - ((float)b2 * grid - 1.0f)) * invg;

      float2 e[8];
      if (l < kDense) {
        // Reference uses stride `res` (not res+1) into the (res+1)^3 table.
        const int base = (b0 * res + b1) * res + b2;
        const float2* __restrict__ tab = tables + kOff[l];
        #pragma unroll
        for (int c = 0; c < 8; ++c) {
          const int idx = base + ((c >> 2) & 1) * res * res
                               + ((c >> 1) & 1) * res + (c & 1);
          e[c] = (l == 0) ? lut0[idx] : tab[idx];   // level 0 from LDS
        }
      } else {
        // h = c0*1 ^ c1*P1 ^ c2*P2, factored per-axis (2 mults / 8 corners)
        const uint32_t hx0 = (uint32_t)b0,       hx1 = hx0 + 1u;
        const uint32_t hy0 = (uint32_t)b1 * P1,  hy1 = hy0 + P1;
        const uint32_t hz0 = (uint32_t)b2 * P2,  hz1 = hz0 + P2;
        const float2* __restrict__ tab = tables + kOff[l];
        #pragma unroll
        for (int c = 0; c < 8; ++c) {
          const uint32_t h = ((c & 4) ? hx1 : hx0) ^
                             ((c & 2) ? hy1 : hy0) ^
                             ((c & 1) ? hz1 : hz0);
          e[c] = tab[h & (uint32_t)(kT - 1)];
        }
      }

      // Trilinear interp, reference combine order (x, then y, then z).
      const float2 c00 = lrp(e[0], e[4], w0);
      const float2 c01 = lrp(e[1], e[5], w0);
      const float2 c10 = lrp(e[2], e[6], w0);
      const float2 c11 = lrp(e[3], e[7], w0);
      const float2 c0  = lrp(c00, c10, w1);
      const float2 c1  = lrp(c01, c11, w1);
      const float2 r   = lrp(c0, c1, w2);
      o[2 * l + 0] = r.x;
      o[2 * l + 1] = r.y;
    }

    // 128B contiguous per point; non-temporal so the 128MB output stream
    // does not evict the L2-resident table.
    v4f* dst = (v4f*)(out + (size_t)p * (2 * kLevels));
    #pragma unroll
    for (int i = 0; i < 8; ++i) {
      v4f v;
      v.x = o[4 * i + 0]; v.y = o[4 * i + 1];
      v.z = o[4 * i + 2]; v.w = o[4 * i + 3];
      __builtin_nontemporal_store(v, dst + i);
    }
  }
}

extern "C" void kernel_launch(void* const* d_in, const int* in_sizes, int n_in,
                              void* d_out, int out_size, void* d_ws, size_t ws_size,
                              hipStream_t stream) {
  const float*  x      = (const float*)d_in[0];
  const float2* tables = (const float2*)d_in[1];
  float*        out    = (float*)d_out;
  const int npts = in_sizes[0] / 3;           // x is (N, 3) float32

  const int threads = 256;                    // 8 wave32 waves per block
  int blocks = 1024;                          // amortizes level-0 LDS staging
  const int maxb = (npts + threads - 1) / threads;
  if (blocks > maxb) blocks = maxb;
  if (blocks < 1) blocks = 1;

  hipLaunchKernelGGL(hashgrid3d_kernel, dim3(blocks), dim3(threads), 0, stream,
                     x, tables, out, npts);
}